// CompressedModel_48619029791190
// MI455X (gfx1250) — compile-verified
//
#include <hip/hip_runtime.h>

#define D      4096
#define BROWS  16
#define LAYERS 4
#define SEGK   8            // split-K factor
#define KC     (D / SEGK)   // 512 K-depth per segment
#define WAVES  8            // waves per block, one 16-col tile each
#define TPB    (WAVES * 32) // 256 threads (8 wave32)

typedef float v2f __attribute__((ext_vector_type(2)));
typedef float v8f __attribute__((ext_vector_type(8)));

// One wave computes a 16(batch) x 16(col) f32 tile over a 512-deep K segment,
// reconstructing W = base + bd*mask on the fly from the streamed operands.
__global__ __launch_bounds__(TPB)
void bitdelta_layer_kernel(const float* __restrict__ x,
                           const float* __restrict__ base,
                           const float* __restrict__ mask,
                           const float* __restrict__ bdp,
                           float* __restrict__ part) {
    __shared__ float xT[KC * BROWS]; // 32 KB: x chunk, transposed [k_local][b]

    const int seg   = blockIdx.x & (SEGK - 1);
    const int tg    = blockIdx.x >> 3;
    const int wave  = threadIdx.x >> 5;  // wave32
    const int lane  = threadIdx.x & 31;
    const int kbase = seg * KC;

    // Cooperative stage of x[:, kbase:kbase+KC] transposed into LDS.
    // Global reads are coalesced along k (row-major x).
    for (int i = threadIdx.x; i < KC * BROWS; i += TPB) {
        int b = i / KC;
        int k = i - b * KC;
        xT[k * BROWS + b] = x[b * D + kbase + k];
    }
    __syncthreads();

    const float bd   = *bdp;                    // uniform scalar load
    const int   tile = tg * WAVES + wave;       // 0..255 -> column tile
    const int   col  = tile * 16 + (lane & 15); // global output column
    const int   hi   = (lane >> 4) << 1;        // 0 or 2 (WMMA f32 A/B K split)

    // B-operand source: rows (kbase+hi), (kbase+hi+1) of W, 16 consecutive cols.
    const float* bp = base + (size_t)(kbase + hi) * D + col;
    const float* mp = mask + (size_t)(kbase + hi) * D + col;
    const float* xp = &xT[hi * BROWS + (lane & 15)];

    v8f c = {};
    #pragma unroll 8
    for (int k0 = 0; k0 < KC; k0 += 4) {
        // Streamed, non-temporal weight operands (zero reuse).
        float b0 = __builtin_nontemporal_load(bp);
        float b1 = __builtin_nontemporal_load(bp + D);
        float m0 = __builtin_nontemporal_load(mp);
        float m1 = __builtin_nontemporal_load(mp + D);

        v2f a, w;
        a.x = xp[0];          // xT[(k0+hi)  *16 + lane&15]  (conflict-free ds)
        a.y = xp[BROWS];      // xT[(k0+hi+1)*16 + lane&15]
        w.x = fmaf(bd, m0, b0);   // fused weight reconstruction
        w.y = fmaf(bd, m1, b1);

        // D(16x16,f32) = A(16x4,f32) x B(4x16,f32) + C
        c = __builtin_amdgcn_wmma_f32_16x16x4_f32(false, a, false, w,
                                                  (short)0, c, false, false);

        bp += (size_t)4 * D;
        mp += (size_t)4 * D;
        xp += 4 * BROWS;
    }

    // C layout: VGPR r -> M=r (lanes 0-15), M=r+8 (lanes 16-31); N = lane&15.
    float*    pout  = part + (size_t)seg * (BROWS * D);
    const int rbase = (lane >> 4) * 8;
    #pragma unroll
    for (int r = 0; r < 8; ++r) {
        pout[(rbase + r) * D + col] = c[r];
    }
}

// Deterministic split-K reduction: out = sum of 8 segment partials.
__global__ __launch_bounds__(256)
void reduce_kernel(const float* __restrict__ part, float* __restrict__ out) {
    int i = blockIdx.x * 256 + threadIdx.x;
    float s = 0.f;
    #pragma unroll
    for (int g = 0; g < SEGK; ++g) s += part[(size_t)g * (BROWS * D) + i];
    out[i] = s;
}

extern "C" void kernel_launch(void* const* d_in, const int* in_sizes, int n_in,
                              void* d_out, int out_size, void* d_ws, size_t ws_size,
                              hipStream_t stream) {
    const float* x   = (const float*)d_in[0];                 // [16,4096]
    const float* bse = (const float*)d_in[1];                 // [4,4096,4096]
    const float* msk = (const float*)d_in[2];                 // [4,4096,4096]
    const float* bd  = (const float*)d_in[3];                 // [4]
    float* out = (float*)d_out;                               // [16,4096]

    float* part = (float*)d_ws;                               // SEGK*16*4096 f32 (2 MB)
    float* xb0  = part + (size_t)SEGK * BROWS * D;            // ping buffer (256 KB)
    float* xb1  = xb0 + (size_t)BROWS * D;                    // pong buffer (256 KB)

    const float* cur = x;
    for (int l = 0; l < LAYERS; ++l) {
        bitdelta_layer_kernel<<<dim3((D / 16 / WAVES) * SEGK), dim3(TPB), 0, stream>>>(
            cur,
            bse + (size_t)l * D * D,
            msk + (size_t)l * D * D,
            bd + l,
            part);
        float* nxt = (l == LAYERS - 1) ? out : ((l & 1) ? xb1 : xb0);
        reduce_kernel<<<dim3((BROWS * D) / 256), dim3(256), 0, stream>>>(part, nxt);
        cur = nxt;
    }
}